// WordStackLSTMCell_63728724738173
// MI455X (gfx1250) — compile-verified
//
#include <hip/hip_runtime.h>
#include <math.h>

// Problem sizes (match the JAX reference)
constexpr int BATCH = 4096;
constexpr int STACK = 64;
constexpr int HID   = 512;
constexpr int INP   = 512;

typedef __attribute__((ext_vector_type(2))) float v2f;
typedef __attribute__((ext_vector_type(8))) float v8f;

__device__ __forceinline__ float sigmoidf_fast(float x) {
    return 1.0f / (1.0f + __expf(-x));
}

// One wave = one 16(batch) x 16(hid) tile, computing all four gate tiles
// (i,f,g,o) via V_WMMA_F32_16X16X4_F32 over K = INP (subword @ W_ih^T) and
// K = HID (gathered h @ W_hh^T), then fused LSTM elementwise + scatter.
//
// Block = 256 threads = 8 waves -> 128 batch rows per block.
// grid.x = BATCH/128, grid.y = HID/16.
__global__ void __launch_bounds__(256)
lstm_wmma_kernel(const float* __restrict__ subword,     // (BATCH, INP)
                 const float* __restrict__ stack_h,     // (BATCH, STACK, HID)
                 const float* __restrict__ stack_c,     // (BATCH, STACK, HID)
                 const int*   __restrict__ pos,         // (BATCH,)
                 const float* __restrict__ W_ih,        // (4*HID, INP) row-major
                 const float* __restrict__ W_hh,        // (4*HID, HID) row-major
                 const float* __restrict__ b_ih,        // (4*HID,)
                 const float* __restrict__ b_hh,        // (4*HID,)
                 float* __restrict__ out_h,             // (BATCH, HID)
                 float* __restrict__ out_c,             // (BATCH, HID)
                 float* __restrict__ out_sh,            // (BATCH, STACK, HID) pre-copied
                 float* __restrict__ out_sc)            // (BATCH, STACK, HID) pre-copied
{
    const int lane  = threadIdx.x & 31;
    const int wave  = threadIdx.x >> 5;
    const int lo    = lane & 15;   // A-frag: M row   | B-frag: N col | D: N col
    const int hi    = lane >> 4;   // K-half selector | D: M = v + 8*hi
    const int rbase = blockIdx.x * 128 + wave * 16;   // batch-row tile base
    const int nbase = blockIdx.y * 16;                // hidden-col tile base

    // ---- A-matrix row pointers for this lane (row = rbase + lo) ----
    const int arow = rbase + lo;
    const float* __restrict__ subw_row = subword + (size_t)arow * INP;
    const int ppos_a = pos[arow];
    const float* __restrict__ h_row =
        stack_h + ((size_t)arow * STACK + ppos_a) * HID;

    // ---- B-matrix (weight) row pointers: column j = g*HID + nbase + lo ----
    const float* __restrict__ wih_row[4];
    const float* __restrict__ whh_row[4];
#pragma unroll
    for (int g = 0; g < 4; ++g) {
        const int j = g * HID + nbase + lo;
        wih_row[g] = W_ih + (size_t)j * INP;
        whh_row[g] = W_hh + (size_t)j * HID;
    }

    v8f acc[4];
#pragma unroll
    for (int g = 0; g < 4; ++g) acc[g] = (v8f)0.0f;

    const int koff = 2 * hi;  // this lane's K sub-offset within a K=4 step

    // ---- GEMM 1: subword @ W_ih^T,  K = INP ----
#pragma unroll 4
    for (int k = 0; k < INP; k += 4) {
        const v2f a = *(const v2f*)(subw_row + k + koff);
#pragma unroll
        for (int g = 0; g < 4; ++g) {
            const v2f w = *(const v2f*)(wih_row[g] + k + koff);
            acc[g] = __builtin_amdgcn_wmma_f32_16x16x4_f32(
                false, a, false, w, (short)0, acc[g], false, false);
        }
    }

    // ---- GEMM 2: gathered h @ W_hh^T,  K = HID ----
#pragma unroll 4
    for (int k = 0; k < HID; k += 4) {
        const v2f a = *(const v2f*)(h_row + k + koff);
#pragma unroll
        for (int g = 0; g < 4; ++g) {
            const v2f w = *(const v2f*)(whh_row[g] + k + koff);
            acc[g] = __builtin_amdgcn_wmma_f32_16x16x4_f32(
                false, a, false, w, (short)0, acc[g], false, false);
        }
    }

    // ---- Bias (per output column j, shared across the 8 rows this lane holds)
    float bias[4];
#pragma unroll
    for (int g = 0; g < 4; ++g) {
        const int j = g * HID + nbase + lo;
        bias[g] = b_ih[j] + b_hh[j];
    }

    // ---- Fused LSTM elementwise + store + scatter ----
    // D layout: element (v, lane) -> m = v + 8*hi, n = lo
    const int col = nbase + lo;
#pragma unroll
    for (int v = 0; v < 8; ++v) {
        const int m    = v + 8 * hi;
        const int brow = rbase + m;
        const int p    = pos[brow];

        const float xi = acc[0][v] + bias[0];
        const float xf = acc[1][v] + bias[1];
        const float xg = acc[2][v] + bias[2];
        const float xo = acc[3][v] + bias[3];

        const float ig = sigmoidf_fast(xi);
        const float fg = sigmoidf_fast(xf);
        const float gg = tanhf(xg);
        const float og = sigmoidf_fast(xo);

        const float c_old = stack_c[((size_t)brow * STACK + p) * HID + col];
        const float c_new = fg * c_old + ig * gg;
        const float h_new = og * tanhf(c_new);

        out_h[(size_t)brow * HID + col] = h_new;
        out_c[(size_t)brow * HID + col] = c_new;

        const size_t sidx = ((size_t)brow * STACK + (p + 1)) * HID + col;
        out_sh[sidx] = h_new;
        out_sc[sidx] = c_new;
    }
}

extern "C" void kernel_launch(void* const* d_in, const int* in_sizes, int n_in,
                              void* d_out, int out_size, void* d_ws, size_t ws_size,
                              hipStream_t stream) {
    const float* subword = (const float*)d_in[0];
    const float* stack_h = (const float*)d_in[1];
    const float* stack_c = (const float*)d_in[2];
    const int*   pos     = (const int*)d_in[3];
    const float* W_ih    = (const float*)d_in[4];
    const float* W_hh    = (const float*)d_in[5];
    const float* b_ih    = (const float*)d_in[6];
    const float* b_hh    = (const float*)d_in[7];

    float* out = (float*)d_out;
    const size_t HC = (size_t)BATCH * HID;          // h_new / c_new element count
    const size_t SC = (size_t)BATCH * STACK * HID;  // stack element count

    float* out_h  = out;                 // h_new
    float* out_c  = out + HC;            // c_new
    float* out_sh = out + 2 * HC;        // new_stack_hidden
    float* out_sc = out + 2 * HC + SC;   // new_stack_cell

    // Pass-through copy of the stacks (dominant cost: ~1 GB of HBM writes).
    // The WMMA kernel below overwrites row pos+1 afterwards (stream-ordered).
    hipMemcpyAsync(out_sh, stack_h, SC * sizeof(float), hipMemcpyDeviceToDevice, stream);
    hipMemcpyAsync(out_sc, stack_c, SC * sizeof(float), hipMemcpyDeviceToDevice, stream);

    dim3 grid(BATCH / 128, HID / 16);   // 32 x 32 blocks, 8 waves each
    lstm_wmma_kernel<<<grid, 256, 0, stream>>>(
        subword, stack_h, stack_c, pos, W_ih, W_hh, b_ih, b_hh,
        out_h, out_c, out_sh, out_sc);
}